// SPDEParameterGNN_41162966564962
// MI455X (gfx1250) — compile-verified
//
#include <hip/hip_runtime.h>
#include <hip/hip_bf16.h>

// ---------------------------------------------------------------------------
// SPDE Parameter GNN for MI455X (gfx1250, wave32).
// Dense GEMMs run on the matrix pipe via V_WMMA_F32_16X16X4_F32 (exact fp32,
// matching the fp32 reference). Edge aggregation uses factorized GCN norm:
//   out[c] = dinv[c] * sum_{(r,c)} (h[r]*dinv[r])   (self-loop added in post)
// so the scatter loop is a pure gather + global_atomic_add_f32 stream.
// ---------------------------------------------------------------------------

typedef __attribute__((ext_vector_type(2))) float v2f;
typedef __attribute__((ext_vector_type(8))) float v8f;

#define N_NODES 100000
#define N_EDGES 1200000
#define D_INP   128
#define H_DIM   64
#define N_OUT   3
// 1/sqrt(1 + 1e-5)
#define BN_SCALE 0.99999500003749971f

// ------------------------------ degree / dinv ------------------------------

__global__ void init_deg_kernel(float* __restrict__ deg, int n) {
    int i = blockIdx.x * blockDim.x + threadIdx.x;
    if (i < n) deg[i] = 1.0f;               // self-loop contribution
}

__global__ void deg_scatter_kernel(const int* __restrict__ col,
                                   float* __restrict__ deg, int e) {
    int i = blockIdx.x * blockDim.x + threadIdx.x;
    if (i < e) atomicAdd(&deg[col[i]], 1.0f);
}

__global__ void make_dinv_kernel(float* __restrict__ deg, int n) {
    int i = blockIdx.x * blockDim.x + threadIdx.x;
    if (i < n) {
        float d = deg[i];
        deg[i] = (d > 0.0f) ? rsqrtf(d) : 0.0f;   // in-place deg -> dinv
    }
}

// ------------------------------ WMMA GEMM ----------------------------------
// Out[n][64] = A[n][K] @ W[64][K]^T (+ bias). One 16x16 tile per wave,
// 4 waves per block cover all 64 output columns of a 16-row stripe.
// f32 A-frag (16x4) and B-frag (4x16): lane L holds K = {k0+2*hi, k0+2*hi+1}
// (hi = L>=16) for row m = L&15 (A) / col n = L&15 (B) -> contiguous float2.

template <int K>
__global__ __launch_bounds__(128)
void gemm_wmma_f32_kernel(const float* __restrict__ A,
                          const float* __restrict__ W,
                          const float* __restrict__ bias,
                          float* __restrict__ out, int n) {
    const int wave = threadIdx.x >> 5;
    const int lane = threadIdx.x & 31;
    const int m    = lane & 15;
    const int hi   = lane >> 4;
    const int row0 = blockIdx.x << 4;
    const int col0 = wave << 4;

    int rowA = row0 + m;
    if (rowA > n - 1) rowA = n - 1;          // n is a multiple of 16; safety clamp
    const float* aRow = A + (size_t)rowA * K;
    const float* bRow = W + (size_t)(col0 + m) * K;

    v8f c = {0.f, 0.f, 0.f, 0.f, 0.f, 0.f, 0.f, 0.f};
#pragma unroll
    for (int kk = 0; kk < K / 4; ++kk) {
        const int k = (kk << 2) + (hi << 1);
        v2f a = *(const v2f*)(aRow + k);
        v2f b = *(const v2f*)(bRow + k);
        c = __builtin_amdgcn_wmma_f32_16x16x4_f32(
                /*neg_a=*/false, a, /*neg_b=*/false, b,
                /*c_mod=*/(short)0, c, /*reuse_a=*/false, /*reuse_b=*/false);
    }

    const float bb = bias ? bias[col0 + m] : 0.0f;
#pragma unroll
    for (int r = 0; r < 8; ++r) {
        const int rr = row0 + r + (hi << 3);
        if (rr < n) out[(size_t)rr * H_DIM + col0 + m] = c[r] + bb;
    }
}

// --------------------- pre-scale (g = t*dinv) + zero acc -------------------
// t and acc share a buffer: consume t, leave zeros for the scatter pass.

__global__ void scale_zero_kernel(float* __restrict__ tacc,
                                  const float* __restrict__ dinv,
                                  float* __restrict__ g, int n) {
    size_t i = ((size_t)blockIdx.x * blockDim.x + threadIdx.x) * 4;
    if (i < (size_t)n * H_DIM) {
        float d = dinv[i >> 6];              // 4-elem chunk never crosses a row
        float4 v = *(const float4*)(tacc + i);
        float4 r = {v.x * d, v.y * d, v.z * d, v.w * d};
        *(float4*)(g + i) = r;
        float4 z = {0.f, 0.f, 0.f, 0.f};
        *(float4*)(tacc + i) = z;
    }
}

// ------------------------------ edge scatter -------------------------------
// 16 threads per edge (4 floats each): coalesced float4 gather of g[row],
// 4x global_atomic_add_f32 into acc[col]. L2-resident, atomic-throughput bound.

__global__ void edge_scatter_kernel(const int* __restrict__ row,
                                    const int* __restrict__ col,
                                    const float* __restrict__ g,
                                    float* __restrict__ acc, int e) {
    long long t = (long long)blockIdx.x * blockDim.x + threadIdx.x;
    int eid = (int)(t >> 4);
    if (eid >= e) return;
    int c = ((int)t & 15) << 2;
    int r = row[eid];
    int d = col[eid];
    float4 v = *(const float4*)(g + (size_t)r * H_DIM + c);
    float* dst = acc + (size_t)d * H_DIM + c;
    atomicAdd(dst + 0, v.x);
    atomicAdd(dst + 1, v.y);
    atomicAdd(dst + 2, v.z);
    atomicAdd(dst + 3, v.w);
}

// --------------- post: self-loop + dinv + bias + BN + ReLU + res -----------

__global__ void post_kernel(const float* __restrict__ acc,
                            const float* __restrict__ g,
                            const float* __restrict__ dinv,
                            const float* __restrict__ conv_b,
                            const float* __restrict__ bn_g,
                            const float* __restrict__ bn_b,
                            float* __restrict__ h, int n, int residual) {
    size_t i = (size_t)blockIdx.x * blockDim.x + threadIdx.x;
    if (i < (size_t)n * H_DIM) {
        int f = (int)(i & 63);
        size_t node = i >> 6;
        float z = (acc[i] + g[i]) * dinv[node] + conv_b[f];   // agg + self + bias
        z = z * (bn_g[f] * BN_SCALE) + bn_b[f];               // batch-norm affine
        z = fmaxf(z, 0.0f);                                   // relu
        if (residual) z += h[i];
        h[i] = z;
    }
}

// ------------------------------ head ---------------------------------------

__global__ void head_kernel(const float* __restrict__ h,
                            const float* __restrict__ x,
                            const float* __restrict__ byp_W,
                            const float* __restrict__ byp_b,
                            const float* __restrict__ head_W,
                            const float* __restrict__ head_b,
                            float* __restrict__ out, int n) {
    int i = blockIdx.x * blockDim.x + threadIdx.x;
    if (i >= n) return;
    const float* xi = x + (size_t)i * D_INP;
    const float* hi = h + (size_t)i * H_DIM;

    float byp[N_OUT];
#pragma unroll
    for (int j = 0; j < N_OUT; ++j) {
        float s = byp_b[j];
        const float* wj = byp_W + j * D_INP;
#pragma unroll 4
        for (int k = 0; k < D_INP; ++k) s += wj[k] * xi[k];
        byp[j] = s;
    }

    float p[N_OUT];
#pragma unroll
    for (int j = 0; j < N_OUT; ++j) {
        float s = head_b[j];
        const float* wj = head_W + j * (H_DIM + N_OUT);
#pragma unroll 4
        for (int k = 0; k < H_DIM; ++k) s += wj[k] * hi[k];
#pragma unroll
        for (int c = 0; c < N_OUT; ++c) s += wj[H_DIM + c] * byp[c];
        p[j] = s;
    }

    float kappa = fminf(fmaxf(p[0] * 5.0f + 2.5f, 0.2f), 10.0f);
    float tau   = fminf(fmaxf(p[2] + 0.5f, 0.05f), 2.0f);
    out[(size_t)i * 3 + 0] = kappa;
    out[(size_t)i * 3 + 1] = p[1];
    out[(size_t)i * 3 + 2] = tau;
}

// ------------------------------ launch -------------------------------------

extern "C" void kernel_launch(void* const* d_in, const int* in_sizes, int n_in,
                              void* d_out, int out_size, void* d_ws, size_t ws_size,
                              hipStream_t stream) {
    const float* x       = (const float*)d_in[0];
    const int*   ei      = (const int*)  d_in[1];
    const float* proj_W  = (const float*)d_in[2];
    const float* proj_b  = (const float*)d_in[3];
    const float* conv_W[3] = {(const float*)d_in[4],  (const float*)d_in[8],  (const float*)d_in[12]};
    const float* conv_b[3] = {(const float*)d_in[5],  (const float*)d_in[9],  (const float*)d_in[13]};
    const float* bn_g[3]   = {(const float*)d_in[6],  (const float*)d_in[10], (const float*)d_in[14]};
    const float* bn_b[3]   = {(const float*)d_in[7],  (const float*)d_in[11], (const float*)d_in[15]};
    const float* byp_W   = (const float*)d_in[16];
    const float* byp_b   = (const float*)d_in[17];
    const float* head_W  = (const float*)d_in[18];
    const float* head_b  = (const float*)d_in[19];
    float* out = (float*)d_out;

    const int n = N_NODES, e = N_EDGES;
    const int* row = ei;
    const int* col = ei + e;

    // workspace layout (floats): dinv[N] | h[N*64] | g[N*64] | t/acc[N*64]
    float* dinv = (float*)d_ws;
    float* h    = dinv + n;
    float* g    = h + (size_t)n * H_DIM;
    float* tacc = g + (size_t)n * H_DIM;

    // degree -> dinv
    init_deg_kernel<<<(n + 255) / 256, 256, 0, stream>>>(dinv, n);
    deg_scatter_kernel<<<(e + 255) / 256, 256, 0, stream>>>(col, dinv, e);
    make_dinv_kernel<<<(n + 255) / 256, 256, 0, stream>>>(dinv, n);

    const int gemm_grid = (n + 15) / 16;          // 6250 blocks, 4 waves each
    const int elem_grid = ((n * H_DIM) + 255) / 256;
    const int vec4_grid = ((n * H_DIM / 4) + 255) / 256;
    const long long sc  = (long long)e * 16;
    const int scat_grid = (int)((sc + 255) / 256);

    // projection: h = x @ proj_W^T + proj_b   (WMMA f32, K=128)
    gemm_wmma_f32_kernel<D_INP><<<gemm_grid, 128, 0, stream>>>(x, proj_W, proj_b, h, n);

    // three GCN layers (residual on first two)
    for (int l = 0; l < 3; ++l) {
        gemm_wmma_f32_kernel<H_DIM><<<gemm_grid, 128, 0, stream>>>(h, conv_W[l], nullptr, tacc, n);
        scale_zero_kernel<<<vec4_grid, 256, 0, stream>>>(tacc, dinv, g, n);
        edge_scatter_kernel<<<scat_grid, 256, 0, stream>>>(row, col, g, tacc, e);
        post_kernel<<<elem_grid, 256, 0, stream>>>(tacc, g, dinv, conv_b[l], bn_g[l], bn_b[l],
                                                   h, n, l < 2 ? 1 : 0);
    }

    // bypass + head + clipping
    head_kernel<<<(n + 255) / 256, 256, 0, stream>>>(h, x, byp_W, byp_b, head_W, head_b, out, n);
}